// AGNNConv_58342835749534
// MI455X (gfx1250) — compile-verified
//
#include <hip/hip_runtime.h>
#include <math.h>

typedef float v2f __attribute__((ext_vector_type(2)));
typedef float v8f __attribute__((ext_vector_type(8)));

#define FD 128  // feature dim (fixed by the reference)

// ---------------------------------------------------------------------------
// Init per-node softmax state; scalar fallback for norm on tail rows (n%16).
// ---------------------------------------------------------------------------
__global__ void init_nodes_kernel(const float* __restrict__ feat,
                                  float* __restrict__ seg_max,
                                  float* __restrict__ seg_sum,
                                  float* __restrict__ invnorm,
                                  int n, int tail_start) {
    int i = blockIdx.x * blockDim.x + threadIdx.x;
    if (i >= n) return;
    seg_max[i] = -INFINITY;
    seg_sum[i] = 0.0f;
    if (i >= tail_start) {
        const float* row = feat + (size_t)i * FD;
        float s = 0.0f;
        #pragma unroll 4
        for (int j = 0; j < FD; ++j) { float v = row[j]; s += v * v; }
        invnorm[i] = 1.0f / fmaxf(sqrtf(s), 1e-12f);
    }
}

// ---------------------------------------------------------------------------
// out = (1 + eps) * feat   (vectorized float4)
// ---------------------------------------------------------------------------
__global__ void out_init_kernel(const float* __restrict__ feat,
                                const float* __restrict__ eps,
                                float* __restrict__ out, int nd4) {
    int i = blockIdx.x * blockDim.x + threadIdx.x;
    if (i >= nd4) return;
    float s = 1.0f + eps[0];
    float4 f = ((const float4*)feat)[i];
    f.x *= s; f.y *= s; f.z *= s; f.w *= s;
    ((float4*)out)[i] = f;
}

// ---------------------------------------------------------------------------
// Squared row norms via V_WMMA_F32_16X16X4_F32: one wave per 16-row tile.
// A = (16x4 chunk of feat) squared elementwise, B = ones(4x16),
// C accumulates over K=128 -> C[m][*] = ||row m||^2 (all columns equal).
// A layout (f32 16x4): lanes 0-15 hold M=lane, K={0,1} in v[0],v[1];
//                      lanes 16-31 hold M=lane-16, K={2,3}.
// C layout: VGPR v, lanes 0-15 -> M=v (N=lane); lanes 16-31 -> M=8+v.
// ---------------------------------------------------------------------------
__global__ void norm_wmma_kernel(const float* __restrict__ feat,
                                 float* __restrict__ invnorm,
                                 int ntiles) {
    int wave = threadIdx.x >> 5;
    int lane = threadIdx.x & 31;
    int tile = blockIdx.x * (blockDim.x >> 5) + wave;
    if (tile >= ntiles) return;

    int row0  = tile << 4;
    int m     = lane & 15;
    int khalf = (lane >> 4) << 1;              // 0 or 2
    const float* base = feat + (size_t)(row0 + m) * FD + khalf;

    v8f c = {};
    v2f ones; ones.x = 1.0f; ones.y = 1.0f;

    #pragma unroll
    for (int k = 0; k < FD; k += 4) {
        v2f a = *(const v2f*)(base + k);       // 8B aligned
        a.x *= a.x; a.y *= a.y;
        c = __builtin_amdgcn_wmma_f32_16x16x4_f32(
                /*neg_a=*/false, a, /*neg_b=*/false, ones,
                /*c_mod=*/(short)0, c, /*reuse_a=*/false, /*reuse_b=*/false);
    }

    // lane 0 owns rows 0..7 (N=0), lane 16 owns rows 8..15 (N=0)
    if (m == 0) {
        int mb = row0 + ((lane >> 4) << 3);
        #pragma unroll
        for (int v = 0; v < 8; ++v)
            invnorm[mb + v] = 1.0f / fmaxf(sqrtf(c[v]), 1e-12f);
    }
}

// ---------------------------------------------------------------------------
// Float atomic max via monotonic int mapping (works for mixed signs):
//   positive floats: signed int compare matches float order
//   negative floats: unsigned compare is reversed -> atomicMin on unsigned
// ---------------------------------------------------------------------------
__device__ __forceinline__ void atomicMaxF32(float* addr, float v) {
    if (v >= 0.0f) atomicMax((int*)addr, __float_as_int(v));
    else           atomicMin((unsigned int*)addr, (unsigned int)__float_as_int(v));
}

__global__ void edge_max_kernel(const float* __restrict__ w,
                                const int* __restrict__ src,
                                const float* __restrict__ beta,
                                float* __restrict__ seg_max, int E) {
    int e = blockIdx.x * blockDim.x + threadIdx.x;
    if (e >= E) return;
    atomicMaxF32(&seg_max[src[e]], beta[0] * w[e]);
}

__global__ void edge_exp_kernel(const float* __restrict__ w,
                                const int* __restrict__ src,
                                const float* __restrict__ beta,
                                const float* __restrict__ seg_max,
                                float* __restrict__ seg_sum,
                                float* __restrict__ ex, int E) {
    int e = blockIdx.x * blockDim.x + threadIdx.x;
    if (e >= E) return;
    int s = src[e];
    float x = __expf(beta[0] * w[e] - seg_max[s]);
    ex[e] = x;
    atomicAdd(&seg_sum[s], x);
}

// ---------------------------------------------------------------------------
// One wave per edge: gather 128-float src row (float4/lane, coalesced 512B),
// scale by p * invnorm[src], atomically scatter-add into out[dst].
// feat/out fit in the 192MB L2 -> gathers and atomics resolve in L2.
// ---------------------------------------------------------------------------
__global__ void scatter_kernel(const float* __restrict__ feat,
                               const int* __restrict__ src,
                               const int* __restrict__ dst,
                               const float* __restrict__ ex,
                               const float* __restrict__ seg_sum,
                               const float* __restrict__ invnorm,
                               float* __restrict__ out, int E) {
    long long tid = (long long)blockIdx.x * blockDim.x + threadIdx.x;
    int e = (int)(tid >> 5);
    if (e >= E) return;
    int lane = (int)(tid & 31);
    int s = src[e], d = dst[e];

    if (lane == 0 && e + 8 < E)  // pre-touch an upcoming source row (L2 prefetch)
        __builtin_prefetch(feat + (size_t)src[e + 8] * FD, 0, 0);

    float coef = (ex[e] / seg_sum[s]) * invnorm[s];
    float4 f = ((const float4*)(feat + (size_t)s * FD))[lane];
    float* o = out + (size_t)d * FD + lane * 4;
    atomicAdd(o + 0, coef * f.x);
    atomicAdd(o + 1, coef * f.y);
    atomicAdd(o + 2, coef * f.z);
    atomicAdd(o + 3, coef * f.w);
}

// ---------------------------------------------------------------------------
extern "C" void kernel_launch(void* const* d_in, const int* in_sizes, int n_in,
                              void* d_out, int out_size, void* d_ws, size_t ws_size,
                              hipStream_t stream) {
    const float* feat = (const float*)d_in[0];
    const float* w    = (const float*)d_in[1];
    const int*   src  = (const int*)d_in[2];
    const int*   dst  = (const int*)d_in[3];
    const float* beta = (const float*)d_in[4];
    const float* eps  = (const float*)d_in[5];
    float* out = (float*)d_out;

    const int n = in_sizes[0] / FD;   // 50000
    const int E = in_sizes[1];        // 640000

    // workspace layout: seg_max[n] | seg_sum[n] | invnorm[n] | ex[E]
    float* seg_max = (float*)d_ws;
    float* seg_sum = seg_max + n;
    float* invnorm = seg_sum + n;
    float* ex      = invnorm + n;

    const int ntiles     = n >> 4;         // 16 rows per wave tile
    const int tail_start = ntiles << 4;

    init_nodes_kernel<<<(n + 255) / 256, 256, 0, stream>>>(
        feat, seg_max, seg_sum, invnorm, n, tail_start);

    const int nd4 = (n * FD) / 4;
    out_init_kernel<<<(nd4 + 255) / 256, 256, 0, stream>>>(feat, eps, out, nd4);

    if (ntiles > 0) {
        const int waves_per_block = 8;   // 256 threads = 8 waves on wave32
        const int blocks = (ntiles + waves_per_block - 1) / waves_per_block;
        norm_wmma_kernel<<<blocks, waves_per_block * 32, 0, stream>>>(
            feat, invnorm, ntiles);
    }

    edge_max_kernel<<<(E + 255) / 256, 256, 0, stream>>>(w, src, beta, seg_max, E);
    edge_exp_kernel<<<(E + 255) / 256, 256, 0, stream>>>(
        w, src, beta, seg_max, seg_sum, ex, E);

    const long long total = (long long)E * 32;
    scatter_kernel<<<(unsigned int)((total + 255) / 256), 256, 0, stream>>>(
        feat, src, dst, ex, seg_sum, invnorm, out, E);
}